// GNNRefiner_12902081757816
// MI455X (gfx1250) — compile-verified
//
#include <hip/hip_runtime.h>
#include <hip/hip_bf16.h>

// ---------------------------------------------------------------------------
// GNN refiner for MI455X (gfx1250, wave32).
//  - All 128-wide GEMMs on V_WMMA_F32_16X16X4_F32 (native f32 tensor op,
//    matches reference precision; GEMM side is ~10 GFLOP).
//  - LDS staging via GLOBAL_LOAD_ASYNC_TO_LDS_B128 (ASYNCcnt) when the
//    builtin exists; register-staged float4 fallback otherwise.
//  - Edge aggregation: wave-per-edge float4 gather + global_atomic_add_f32;
//    the 51MB xw/agg working set lives in the 192MB L2.
// ---------------------------------------------------------------------------

typedef __attribute__((ext_vector_type(2))) float v2f;
typedef __attribute__((ext_vector_type(8))) float v8f;

#if defined(__AMDGCN__) && __has_builtin(__builtin_amdgcn_global_load_async_to_lds_b128)
#define HAVE_ASYNC_LDS 1
// builtin prototype (from hipcc diagnostic): (v4i as1*, v4i as3*, imm, imm)
typedef int v4i_vs __attribute__((vector_size(16)));
typedef __attribute__((address_space(1))) v4i_vs* as1_v4i_p;
typedef __attribute__((address_space(3))) v4i_vs* as3_v4i_p;
#else
#define HAVE_ASYNC_LDS 0
#endif

__device__ __forceinline__ void atomic_add_f32(float* p, float v) {
    __hip_atomic_fetch_add(p, v, __ATOMIC_RELAXED, __HIP_MEMORY_SCOPE_AGENT);
}

// copy 16 bytes global -> LDS
__device__ __forceinline__ void stage16(const float* __restrict__ g, float* l) {
#if HAVE_ASYNC_LDS
    __builtin_amdgcn_global_load_async_to_lds_b128(
        (as1_v4i_p)(g), (as3_v4i_p)(l), 0, 0);
#else
    *(float4*)l = *(const float4*)g;
#endif
}

__device__ __forceinline__ void stage_fence() {
#if HAVE_ASYNC_LDS
#if __has_builtin(__builtin_amdgcn_s_wait_asynccnt)
    __builtin_amdgcn_s_wait_asynccnt(0);
#else
    asm volatile("s_wait_asynccnt 0x0" ::: "memory");
#endif
#endif
}

// ------------------------- feature construction ----------------------------
// raw[i][0..6] = geom, raw[i][7..22] = emb[labels[i]], raw[i][23..31] = 0
__global__ __launch_bounds__(256) void k_build_raw(
    const float* __restrict__ boxes, const float* __restrict__ scores,
    const int* __restrict__ labels, const int* __restrict__ Hp,
    const int* __restrict__ Wp, const float* __restrict__ emb,
    float* __restrict__ raw, int n)
{
    int t = blockIdx.x * 256 + threadIdx.x;
    int i = t >> 5;
    int c = t & 31;
    if (i >= n) return;
    float Hf = (float)Hp[0];
    float Wf = (float)Wp[0];
    float val = 0.0f;
    if (c < 7) {
        float x1 = boxes[i * 4 + 0], y1 = boxes[i * 4 + 1];
        float x2 = boxes[i * 4 + 2], y2 = boxes[i * 4 + 3];
        float w  = fmaxf(x2 - x1, 1.0f);
        float h  = fmaxf(y2 - y1, 1.0f);
        float cx = (x1 + x2) * 0.5f;
        float cy = (y1 + y2) * 0.5f;
        switch (c) {
            case 0: val = cx / Wf; break;
            case 1: val = cy / Hf; break;
            case 2: val = w / Wf; break;
            case 3: val = h / Hf; break;
            case 4: val = (w / Wf) * (h / Hf); break;
            case 5: val = w / (h + 1e-6f); break;
            default: val = scores[i]; break;
        }
    } else if (c < 23) {
        val = emb[labels[i] * 16 + (c - 7)];
    }
    raw[(size_t)i * 32 + c] = val;
}

// pad W1 (23x128) -> W1p (32x128), zero rows 23..31
__global__ __launch_bounds__(256) void k_pad_w1(const float* __restrict__ W1,
                                                float* __restrict__ W1p)
{
    int t = blockIdx.x * 256 + threadIdx.x;   // 0..4095
    int k = t >> 7;
    int c = t & 127;
    W1p[t] = (k < 23) ? W1[k * 128 + c] : 0.0f;
}

// ------------------------------ WMMA GEMM ----------------------------------
// out[M x 128] = epilogue(X[M x K] @ W[K x 128] (+bias)(relu)),  K in {32,128}
// Block = 256 threads = 8 waves, computes a 64-row x 128-col macro tile.
// Wave wv owns cols [16wv,16wv+16) and 4 row tiles (4 v8f accumulators),
// so every staged B fragment feeds 4 WMMAs.
#define SX_STRIDE 36     // 32 + pad; 36*4B = 144B (16B multiple)
#define SW_STRIDE 136    // 128 + pad; 136*4B = 544B (16B multiple); halves of
                         // the B fetch land in disjoint LDS bank groups

__global__ __launch_bounds__(256) void k_gemm_wmma(
    const float* __restrict__ X, const float* __restrict__ W,
    const float* __restrict__ bias, float* __restrict__ out,
    int M, int K, int relu)
{
    __shared__ float sx[64 * SX_STRIDE];   // 64 rows x 32 K      (9.2 KB)
    __shared__ float sw[32 * SW_STRIDE];   // 32 K   x 128 cols  (17.4 KB)

    int t    = threadIdx.x;
    int r0   = blockIdx.x * 64;
    int wv   = t >> 5;            // wave id 0..7 -> output column tile
    int lane = t & 31;
    int half = lane >> 4;         // 0: K pair {0,1}, 1: K pair {2,3}
    int m    = lane & 15;         // row (A) / column (B,D) within tile

    v8f acc[4] = {v8f{}, v8f{}, v8f{}, v8f{}};

    for (int k0 = 0; k0 < K; k0 += 32) {
        __syncthreads();
        // stage X strip 64x32: thread t -> row t>>2, 8 floats at col (t&3)*8
        {
            int r  = t >> 2;
            int c0 = (t & 3) * 8;
            int rr = r0 + r; if (rr >= M) rr = M - 1;
            const float* gp = X + (size_t)rr * K + k0 + c0;
            float* lp = &sx[r * SX_STRIDE + c0];
            stage16(gp, lp);
            stage16(gp + 4, lp + 4);
        }
        // stage W chunk 32x128: thread t -> row t>>3, 16 floats at col (t&7)*16
        {
            int r  = t >> 3;
            int c0 = (t & 7) * 16;
            const float* gp = W + (size_t)(k0 + r) * 128 + c0;
            float* lp = &sw[r * SW_STRIDE + c0];
            stage16(gp,      lp);
            stage16(gp + 4,  lp + 4);
            stage16(gp + 8,  lp + 8);
            stage16(gp + 12, lp + 12);
        }
        stage_fence();
        __syncthreads();

#pragma unroll
        for (int kk = 0; kk < 32; kk += 4) {
            int ka = kk + 2 * half;
            // B 4x16: row = K, col striped across lanes
            v2f b;
            b.x = sw[ka * SW_STRIDE + wv * 16 + m];
            b.y = sw[(ka + 1) * SW_STRIDE + wv * 16 + m];
            // A 16x4 per row tile: lanes0-15 -> K{0,1}, lanes16-31 -> K{2,3}
#pragma unroll
            for (int rt = 0; rt < 4; ++rt) {
                v2f a;
                a.x = sx[(rt * 16 + m) * SX_STRIDE + ka];
                a.y = sx[(rt * 16 + m) * SX_STRIDE + ka + 1];
                acc[rt] = __builtin_amdgcn_wmma_f32_16x16x4_f32(
                    false, a, false, b, (short)0, acc[rt], false, false);
            }
        }
    }

    int col = wv * 16 + m;
    float bv = bias ? bias[col] : 0.0f;
#pragma unroll
    for (int rt = 0; rt < 4; ++rt) {
#pragma unroll
        for (int v = 0; v < 8; ++v) {
            int row = r0 + rt * 16 + v + 8 * half;  // D: VGPR v -> M=v / v+8
            if (row < M) {
                float val = acc[rt][v] + bv;
                if (relu) val = fmaxf(val, 0.0f);
                out[(size_t)row * 128 + col] = val;
            }
        }
    }
}

// --------------------------- GCN normalization -----------------------------
__global__ __launch_bounds__(256) void k_deg_init(float* __restrict__ deg, int n)
{
    int i = blockIdx.x * 256 + threadIdx.x;
    if (i < n) deg[i] = 1.0f;   // self-loop weight
}

__global__ __launch_bounds__(256) void k_deg_scatter(
    const int* __restrict__ dst, const float* __restrict__ ew,
    float* __restrict__ deg, int e)
{
    int i = blockIdx.x * 256 + threadIdx.x;
    if (i < e) atomic_add_f32(deg + dst[i], ew[i]);
}

__global__ __launch_bounds__(256) void k_dis(const float* __restrict__ deg,
                                             float* __restrict__ dis, int n)
{
    int i = blockIdx.x * 256 + threadIdx.x;
    if (i < n) {
        float d = deg[i];
        dis[i] = (d > 0.0f) ? rsqrtf(fmaxf(d, 1e-12f)) : 0.0f;
    }
}

// agg[i][c] = xw[i][c] * dis[i]^2   (self-loop term; initializes agg)
__global__ __launch_bounds__(256) void k_agg_init(
    const float* __restrict__ xw, const float* __restrict__ dis,
    float* __restrict__ agg, int n)
{
    int t = blockIdx.x * 256 + threadIdx.x;   // one float4 per thread
    if (t < n * 32) {
        int i = t >> 5;
        float di = dis[i];
        float s = di * di;
        float4 v = ((const float4*)xw)[t];
        v.x *= s; v.y *= s; v.z *= s; v.w *= s;
        ((float4*)agg)[t] = v;
    }
}

// one wave per edge; lane handles 4 channels (float4 gather + 4 f32 atomics)
__global__ __launch_bounds__(256) void k_edge_scatter(
    const int* __restrict__ src, const int* __restrict__ dst,
    const float* __restrict__ ew, const float* __restrict__ dis,
    const float* __restrict__ xw, float* __restrict__ agg, int e)
{
    int wave = (blockIdx.x * 256 + threadIdx.x) >> 5;
    if (wave >= e) return;
    int lane = threadIdx.x & 31;
    int s = src[wave];
    int d = dst[wave];
    float norm = dis[s] * ew[wave] * dis[d];
    float4 v = ((const float4*)(xw + (size_t)s * 128))[lane];
    float* ap = agg + (size_t)d * 128 + lane * 4;
    atomic_add_f32(ap + 0, v.x * norm);
    atomic_add_f32(ap + 1, v.y * norm);
    atomic_add_f32(ap + 2, v.z * norm);
    atomic_add_f32(ap + 3, v.w * norm);
}

// x = relu(agg + bg) + x
__global__ __launch_bounds__(256) void k_finish(
    const float* __restrict__ agg, const float* __restrict__ bg,
    float* __restrict__ x, int n)
{
    int t = blockIdx.x * 256 + threadIdx.x;   // one float4 per thread
    if (t < n * 32) {
        int c4 = t & 31;
        float4 a = ((const float4*)agg)[t];
        float4 b = ((const float4*)bg)[c4];
        float4 xv = ((const float4*)x)[t];
        xv.x += fmaxf(a.x + b.x, 0.0f);
        xv.y += fmaxf(a.y + b.y, 0.0f);
        xv.z += fmaxf(a.z + b.z, 0.0f);
        xv.w += fmaxf(a.w + b.w, 0.0f);
        ((float4*)x)[t] = xv;
    }
}

// --------------------------------- head ------------------------------------
__global__ __launch_bounds__(256) void k_head(
    const float* __restrict__ x, const float* __restrict__ hW,
    const float* __restrict__ hb, float* __restrict__ out, int n)
{
    __shared__ float shw[512];
    __shared__ float shb[4];
    for (int idx = threadIdx.x; idx < 512; idx += 256) shw[idx] = hW[idx];
    if (threadIdx.x < 4) shb[threadIdx.x] = hb[threadIdx.x];
    __syncthreads();

    int i = blockIdx.x * 256 + threadIdx.x;
    if (i >= n) return;
    const float4* xr = (const float4*)(x + (size_t)i * 128);
    float s0 = 0.f, s1 = 0.f, s2 = 0.f, s3 = 0.f;
#pragma unroll 8
    for (int c4 = 0; c4 < 32; ++c4) {
        float4 v = xr[c4];
        const float* w = &shw[c4 * 16];   // 4 channels x 4 outputs
        s0 += v.x * w[0] + v.y * w[4] + v.z * w[8]  + v.w * w[12];
        s1 += v.x * w[1] + v.y * w[5] + v.z * w[9]  + v.w * w[13];
        s2 += v.x * w[2] + v.y * w[6] + v.z * w[10] + v.w * w[14];
        s3 += v.x * w[3] + v.y * w[7] + v.z * w[11] + v.w * w[15];
    }
    out[(size_t)i * 4 + 0] = s0 + shb[0];
    out[(size_t)i * 4 + 1] = s1 + shb[1];
    out[(size_t)i * 4 + 2] = s2 + shb[2];
    out[(size_t)i * 4 + 3] = s3 + shb[3];
}

// ------------------------------- launcher ----------------------------------
extern "C" void kernel_launch(void* const* d_in, const int* in_sizes, int n_in,
                              void* d_out, int out_size, void* d_ws, size_t ws_size,
                              hipStream_t stream)
{
    const float* boxes  = (const float*)d_in[0];
    const float* scores = (const float*)d_in[1];
    const int*   labels = (const int*)d_in[2];
    const int*   Hp     = (const int*)d_in[3];
    const int*   Wp     = (const int*)d_in[4];
    const int*   eidx   = (const int*)d_in[5];
    const float* ew     = (const float*)d_in[6];
    const float* emb    = (const float*)d_in[7];
    const float* W1     = (const float*)d_in[8];
    const float* b1     = (const float*)d_in[9];
    const float* W2     = (const float*)d_in[10];
    const float* b2     = (const float*)d_in[11];
    const float* gW[3]  = {(const float*)d_in[12], (const float*)d_in[14], (const float*)d_in[16]};
    const float* gb[3]  = {(const float*)d_in[13], (const float*)d_in[15], (const float*)d_in[17]};
    const float* hW     = (const float*)d_in[18];
    const float* hb     = (const float*)d_in[19];
    float* out = (float*)d_out;

    const int N = in_sizes[1];          // scores
    const int E = in_sizes[6];          // edge_weight
    const int* src = eidx;
    const int* dst = eidx + E;

    // scratch layout (floats); needs ~169 MB of ws
    float* ws  = (float*)d_ws;
    size_t off = 0;
    float* raw  = ws + off; off += (size_t)N * 32;
    float* W1p  = ws + off; off += 32 * 128;
    float* deg  = ws + off; off += (size_t)N;
    float* dis  = ws + off; off += (size_t)N;
    float* bufA = ws + off; off += (size_t)N * 128;   // x1 / xw
    float* bufB = ws + off; off += (size_t)N * 128;   // x (running state)
    float* bufC = ws + off; off += (size_t)N * 128;   // agg

    int gN    = (N + 255) / 256;
    int gN4   = (N * 32 + 255) / 256;   // N*128 floats as float4
    int gTile = (N + 63) / 64;

    // node features -> x
    k_build_raw<<<(N * 32 + 255) / 256, 256, 0, stream>>>(boxes, scores, labels,
                                                          Hp, Wp, emb, raw, N);
    k_pad_w1<<<16, 256, 0, stream>>>(W1, W1p);
    k_gemm_wmma<<<gTile, 256, 0, stream>>>(raw,  W1p, b1, bufA, N, 32, 1);
    k_gemm_wmma<<<gTile, 256, 0, stream>>>(bufA, W2,  b2, bufB, N, 128, 1);

    // symmetric normalization coefficients (shared by all layers)
    k_deg_init<<<gN, 256, 0, stream>>>(deg, N);
    k_deg_scatter<<<(E + 255) / 256, 256, 0, stream>>>(dst, ew, deg, E);
    k_dis<<<gN, 256, 0, stream>>>(deg, dis, N);

    // three GCN layers with residual
    for (int l = 0; l < 3; ++l) {
        k_gemm_wmma<<<gTile, 256, 0, stream>>>(bufB, gW[l], nullptr, bufA, N, 128, 0);
        k_agg_init<<<gN4, 256, 0, stream>>>(bufA, dis, bufC, N);
        k_edge_scatter<<<(E + 7) / 8, 256, 0, stream>>>(src, dst, ew, dis, bufA, bufC, E);
        k_finish<<<gN4, 256, 0, stream>>>(bufC, gb[l], bufB, N);
    }

    // head
    k_head<<<gN, 256, 0, stream>>>(bufB, hW, hb, out, N);
}